// MessagePassing_73589969649752
// MI455X (gfx1250) — compile-verified
//
#include <hip/hip_runtime.h>
#include <stdint.h>

typedef __attribute__((ext_vector_type(2))) float v2f;
typedef __attribute__((ext_vector_type(8))) float v8f;

#define DFEAT 64
#define XPAD  68    // A-tile row stride (floats): bank = (4*l16 + k) % 64 -> conflict-free
#define WPSTR 96    // paired-W stride (floats): 96 % 64 == 32 -> halves on disjoint banks
#define WAVES 4     // 128-thread blocks
#define NEG_SLOPE 0.01f

// ---- CDNA5 async global->LDS copy (16B per lane per op), tracked by ASYNCcnt ----
__device__ __forceinline__ void async_copy_b128(uint32_t lds_off, const void* gaddr) {
  asm volatile("global_load_async_to_lds_b128 %0, %1, off"
               :: "v"(lds_off), "v"(gaddr) : "memory");
}
__device__ __forceinline__ void wait_async0() {
  asm volatile("s_wait_asynccnt 0x0" ::: "memory");
}
__device__ __forceinline__ uint32_t lds_addr_of(const void* p) {
  // flat LDS aperture: addr[31:0] is the wave-relative LDS byte address
  return (uint32_t)(uintptr_t)p;
}
// compiler-only memory barrier; HW keeps same-wave LDS ops in order (DScnt in-order)
__device__ __forceinline__ void wave_mem_fence() { asm volatile("" ::: "memory"); }

// Stage one 32x64 row-major tile (8 KB) into LDS with XPAD row padding, async.
__device__ __forceinline__ void async_stage_tile32(float* dst, const float* src, int lane) {
  uint32_t lds = lds_addr_of(dst) + (uint32_t)(((lane >> 4) * XPAD + (lane & 15) * 4) * 4);
  const char* g = (const char*)src + lane * 16;
#pragma unroll
  for (int it = 0; it < 16; ++it) {            // 512 x b128 chunks over 32 lanes
    async_copy_b128(lds, g);
    lds += 2 * XPAD * 4;                       // +2 rows
    g   += 512;                                // +2 rows in memory
  }
}

// Weights -> paired-interleaved LDS: pair p holds rows 2p,2p+1; element n at [p*WPSTR + 2n (+ r&1)]
__device__ __forceinline__ void load_w_pairs(float* sWp, const float* __restrict__ W,
                                             int tid, int nthreads) {
  for (int i = tid; i < 64 * 64; i += nthreads) {
    int r = i >> 6, c = i & 63;                // coalesced global read
    sWp[(r >> 1) * WPSTR + 2 * c + (r & 1)] = W[i];
  }
}

// One wave: C[32x64] += A[32x64] @ B[64x64] via V_WMMA_F32_16X16X4_F32.
// A frag: ds_load_b64 (K-pair contiguous). B frag: single ds_load_b64 from paired layout.
__device__ __forceinline__ void wave_gemm32(const float* __restrict__ aLds,
                                            const float* __restrict__ wPair,
                                            v8f acc[8], int l16, int half) {
#pragma unroll
  for (int k0 = 0; k0 < DFEAT; k0 += 4) {
    const int P = (k0 >> 1) + half;            // pair index: rows 2P, 2P+1
    v2f a0, a1;
    const float* ap0 = aLds + l16 * XPAD + k0 + 2 * half;
    a0.x = ap0[0]; a0.y = ap0[1];              // -> ds_load_b64
    const float* ap1 = ap0 + 16 * XPAD;
    a1.x = ap1[0]; a1.y = ap1[1];
#pragma unroll
    for (int t = 0; t < 4; ++t) {
      v2f b;
      const float* bp = wPair + P * WPSTR + 2 * (t * 16 + l16);
      b.x = bp[0]; b.y = bp[1];                // -> single ds_load_b64, even VGPR pair
      acc[t]     = __builtin_amdgcn_wmma_f32_16x16x4_f32(false, a0, false, b,
                                                         (short)0, acc[t], false, false);
      acc[4 + t] = __builtin_amdgcn_wmma_f32_16x16x4_f32(false, a1, false, b,
                                                         (short)0, acc[4 + t], false, false);
    }
  }
}

// ---------------- Kernel 1: m = nodes @ W_node  (m lives in d_ws, L2-resident) -------------
__global__ void __launch_bounds__(WAVES * 32)
node_proj_kernel(const float* __restrict__ nodes, const float* __restrict__ Wn,
                 float* __restrict__ m, int nTiles) {
  __shared__ float sW[32 * WPSTR];             // 12 KB paired W_node
  __shared__ float sA[WAVES][32 * XPAD];       // 34.8 KB staging

  const int tid  = threadIdx.x;
  const int lane = tid & 31;
  const int wave = tid >> 5;
  const int l16  = lane & 15;
  const int half = lane >> 4;

  load_w_pairs(sW, Wn, tid, WAVES * 32);

  const int tile = blockIdx.x * WAVES + wave;  // wave-uniform
  const bool active = tile < nTiles;

  if (active)
    async_stage_tile32(sA[wave], nodes + (size_t)tile * 32 * DFEAT, lane);
  wait_async0();
  __syncthreads();

  if (active) {
    v8f acc[8] = {};
    wave_gemm32(sA[wave], sW, acc, l16, half);
    float* mrow = m + (size_t)tile * 32 * DFEAT;
#pragma unroll
    for (int mt = 0; mt < 2; ++mt)
#pragma unroll
      for (int r = 0; r < 8; ++r) {
        const int row = mt * 16 + r + 8 * half;
#pragma unroll
        for (int t = 0; t < 4; ++t)
          mrow[row * DFEAT + t * 16 + l16] = acc[mt * 4 + t][r];
      }
  }
}

// ------- Kernel 2: fused edge MLP (2 GEMM + bias + leaky) + gather(m)*e + scatter-add ------
__global__ void __launch_bounds__(WAVES * 32)
edge_mp_kernel(const float* __restrict__ edges,
               const int* __restrict__ seg,
               const int* __restrict__ gidx,
               const float* __restrict__ We1, const float* __restrict__ be1,
               const float* __restrict__ We2, const float* __restrict__ be2,
               const float* __restrict__ m,
               float* __restrict__ out, int nTiles) {
  __shared__ float sW1[32 * WPSTR];            // 12 KB
  __shared__ float sW2[32 * WPSTR];            // 12 KB
  __shared__ float sX[WAVES][32 * XPAD];       // 34.8 KB: edge tile, then X1 tile
  __shared__ int   sIdx[WAVES][32], sSeg[WAVES][32];

  const int tid  = threadIdx.x;
  const int lane = tid & 31;
  const int wave = tid >> 5;
  const int l16  = lane & 15;
  const int half = lane >> 4;

  load_w_pairs(sW1, We1, tid, WAVES * 32);
  load_w_pairs(sW2, We2, tid, WAVES * 32);

  const int tile = blockIdx.x * WAVES + wave;  // wave-uniform
  const bool active = tile < nTiles;

  // biases live in registers: value depends only on n = t*16 + l16 (broadcast loads)
  float b1v[4], b2v[4];
#pragma unroll
  for (int t = 0; t < 4; ++t) { b1v[t] = be1[t * 16 + l16]; b2v[t] = be2[t * 16 + l16]; }

  if (active) {
    async_stage_tile32(sX[wave], edges + (size_t)tile * 32 * DFEAT, lane);
    sIdx[wave][lane] = gidx[tile * 32 + lane];
    sSeg[wave][lane] = seg[tile * 32 + lane];
  }
  wait_async0();
  __syncthreads();

  if (active) {
    v8f acc[8] = {};
    wave_gemm32(sX[wave], sW1, acc, l16, half);       // X1 = edges @ W_e1
#pragma unroll
    for (int i = 0; i < 8; ++i) {
      const float b = b1v[i & 3];
#pragma unroll
      for (int r = 0; r < 8; ++r) {
        float v = acc[i][r] + b;
        acc[i][r] = fmaxf(v, v * NEG_SLOPE);          // leaky relu
      }
    }
    wave_mem_fence();                                  // don't hoist stores above GEMM1 reads
#pragma unroll
    for (int mt = 0; mt < 2; ++mt)
#pragma unroll
      for (int r = 0; r < 8; ++r) {
        const int row = mt * 16 + r + 8 * half;
#pragma unroll
        for (int t = 0; t < 4; ++t)
          sX[wave][row * XPAD + t * 16 + l16] = acc[mt * 4 + t][r];
      }
    wave_mem_fence();                                  // same-wave LDS is in-order in HW

    v8f acc2[8] = {};
    wave_gemm32(sX[wave], sW2, acc2, l16, half);       // e = X1 @ W_e2
#pragma unroll
    for (int i = 0; i < 8; ++i) {
      const float b = b2v[i & 3];
#pragma unroll
      for (int r = 0; r < 8; ++r) {
        float v = acc2[i][r] + b;
        acc2[i][r] = fmaxf(v, v * NEG_SLOPE);
      }
    }

    // gather m[index] (L2-resident), multiply, scatter-add into out[seg]
#pragma unroll
    for (int mt = 0; mt < 2; ++mt)
#pragma unroll
      for (int r = 0; r < 8; ++r) {
        const int row = mt * 16 + r + 8 * half;
        const int g = sIdx[wave][row];
        const int s = sSeg[wave][row];
        const float* mrow = m + (size_t)g * DFEAT;
        float* orow = out + (size_t)s * DFEAT;
#pragma unroll
        for (int t = 0; t < 4; ++t) {
          const float mv = mrow[t * 16 + l16];
          atomicAdd(orow + t * 16 + l16, acc2[mt * 4 + t][r] * mv);
        }
      }
  }
}

extern "C" void kernel_launch(void* const* d_in, const int* in_sizes, int n_in,
                              void* d_out, int out_size, void* d_ws, size_t ws_size,
                              hipStream_t stream) {
  const float* nodes = (const float*)d_in[0];
  const float* edges = (const float*)d_in[1];
  const int*   seg   = (const int*)d_in[2];
  const int*   gidx  = (const int*)d_in[3];
  const float* Wn    = (const float*)d_in[4];
  const float* We1   = (const float*)d_in[5];
  const float* be1   = (const float*)d_in[6];
  const float* We2   = (const float*)d_in[7];
  const float* be2   = (const float*)d_in[8];
  float* out = (float*)d_out;
  float* m   = (float*)d_ws;                 // 100000*64*4 = 25.6 MB (fits L2: 192 MB)

  const int nNodes = in_sizes[0] / DFEAT;    // 100000 -> 3125 tiles of 32
  const int nEdges = in_sizes[2];            // 1000000 -> 31250 tiles of 32
  const int nodeTiles = (nNodes + 31) / 32;
  const int edgeTiles = (nEdges + 31) / 32;

  hipMemsetAsync(d_out, 0, (size_t)out_size * sizeof(float), stream);

  const int nb1 = (nodeTiles + WAVES - 1) / WAVES;
  node_proj_kernel<<<nb1, WAVES * 32, 0, stream>>>(nodes, Wn, m, nodeTiles);

  const int nb2 = (edgeTiles + WAVES - 1) / WAVES;
  edge_mp_kernel<<<nb2, WAVES * 32, 0, stream>>>(edges, seg, gidx, We1, be1,
                                                 We2, be2, m, out, edgeTiles);
}